// DynamicGraphAttention_59863254172140
// MI455X (gfx1250) — compile-verified
//
#include <hip/hip_runtime.h>
#include <hip/hip_bf16.h>

typedef __attribute__((ext_vector_type(2))) float v2f;
typedef __attribute__((ext_vector_type(4))) float v4f;
typedef __attribute__((ext_vector_type(8))) float v8f;

#define B_ 4
#define N_ 8192
#define C_ 256
#define K_ 16
#define QPB 4            // queries per block in the neighbor-GEMM kernel
#define MROWS 32         // query rows per block in the T-GEMM kernel

// ---------------------------------------------------------------------------
// Kernel 1: streaming KNN (top-16 smallest squared distances per query).
// One thread per query; position tiles staged in LDS; per-thread sorted
// (descending) 16-entry register list, single bubble pass on insert.
// ---------------------------------------------------------------------------
__global__ __launch_bounds__(256) void knn_kernel(const float* __restrict__ pos,
                                                  int* __restrict__ out_idx) {
    const int TILE = 256;
    int b = blockIdx.x / (N_ / 256);
    int n = (blockIdx.x % (N_ / 256)) * 256 + threadIdx.x;
    const float* bp = pos + (size_t)b * N_ * 3;

    float qx = bp[(size_t)n * 3 + 0];
    float qy = bp[(size_t)n * 3 + 1];
    float qz = bp[(size_t)n * 3 + 2];

    float dist[K_];
    int   ind[K_];
#pragma unroll
    for (int i = 0; i < K_; ++i) { dist[i] = 3.4e38f; ind[i] = 0; }

    __shared__ float sx[TILE], sy[TILE], sz[TILE];

    for (int t0 = 0; t0 < N_; t0 += TILE) {
        __syncthreads();
        int m = t0 + threadIdx.x;
        sx[threadIdx.x] = bp[(size_t)m * 3 + 0];
        sy[threadIdx.x] = bp[(size_t)m * 3 + 1];
        sz[threadIdx.x] = bp[(size_t)m * 3 + 2];
        __syncthreads();

        for (int j = 0; j < TILE; ++j) {
            float dx = qx - sx[j];
            float dy = qy - sy[j];
            float dz = qz - sz[j];
            float d = fmaf(dx, dx, fmaf(dy, dy, dz * dz));
            if (d < dist[0]) {           // dist[0] is current worst (largest)
                dist[0] = d; ind[0] = t0 + j;
#pragma unroll
                for (int i = 0; i < K_ - 1; ++i) {   // restore descending order
                    if (dist[i] < dist[i + 1]) {
                        float td = dist[i]; dist[i] = dist[i + 1]; dist[i + 1] = td;
                        int   ti = ind[i];  ind[i]  = ind[i + 1];  ind[i + 1]  = ti;
                    }
                }
            }
        }
    }

    int* op = out_idx + ((size_t)b * N_ + n) * K_;
#pragma unroll
    for (int i = 0; i < K_; ++i) op[i] = ind[i];
}

// ---------------------------------------------------------------------------
// Kernel 2: T[r, c] = Q[r, :] @ (W1 - W0)[:, c] + bias[c]  for all 32768 rows.
// Dense WMMA GEMM: block = 32 rows (2 M-tiles) x 256 cols; 8 waves x 2 column
// tiles x 2 M-tiles. Q block staged in LDS; Wd fragments built on the fly
// with one packed f32 subtract (W1 row is +65536 floats from W0 row).
// ---------------------------------------------------------------------------
__global__ __launch_bounds__(256) void tgemm_kernel(const float* __restrict__ q,
                                                    const float* __restrict__ W,
                                                    const float* __restrict__ bias,
                                                    float* __restrict__ T) {
    int mbase = blockIdx.x * MROWS;
    int tid  = threadIdx.x;
    int lane = tid & 31;
    int wave = tid >> 5;
    int l16  = lane & 15;
    int roff = (lane >> 4) << 1;       // K offset 0 (lanes 0-15) or 2 (lanes 16-31)
    int ct0  = wave * 2;               // first of this wave's 2 column tiles

    __shared__ float qtile[MROWS * C_];          // 32 KB

    // coalesced stage of the 32x256 Q block
    const v4f* qsrc = (const v4f*)(q + (size_t)mbase * C_);
    v4f* qdst = (v4f*)qtile;
#pragma unroll
    for (int i = 0; i < (MROWS * C_ / 4) / 256; ++i)
        qdst[tid + i * 256] = qsrc[tid + i * 256];
    __syncthreads();

    v8f acc[2][2];                     // [m-tile][col-tile]
#pragma unroll
    for (int mt = 0; mt < 2; ++mt)
#pragma unroll
        for (int t = 0; t < 2; ++t)
            acc[mt][t] = (v8f){0.f, 0.f, 0.f, 0.f, 0.f, 0.f, 0.f, 0.f};

    const int W1OFF = C_ * C_;         // W1 starts 65536 floats after W0

    for (int kk = 0; kk < C_ / 4; ++kk) {
        int krow = kk * 4 + roff;
        const float* wp = W + (size_t)krow * C_ + ct0 * 16 + l16;
        v2f B0, B1;                    // Wd = W1 - W0 fragments
        B0.x = wp[W1OFF]           - wp[0];
        B0.y = wp[W1OFF + C_]      - wp[C_];
        B1.x = wp[W1OFF + 16]      - wp[16];
        B1.y = wp[W1OFF + C_ + 16] - wp[C_ + 16];
#pragma unroll
        for (int mt = 0; mt < 2; ++mt) {
            v2f A = *(const v2f*)(qtile + (mt * 16 + l16) * C_ + kk * 4 + roff);
            acc[mt][0] = __builtin_amdgcn_wmma_f32_16x16x4_f32(
                false, A, false, B0, (short)0, acc[mt][0], false, false);
            acc[mt][1] = __builtin_amdgcn_wmma_f32_16x16x4_f32(
                false, A, false, B1, (short)0, acc[mt][1], false, false);
        }
    }

    // store T = acc + bias.  D layout: VGPR r -> row (mt*16 + r) for lanes 0-15,
    // row (mt*16 + 8 + r) for lanes 16-31; column = ct*16 + l16.
#pragma unroll
    for (int mt = 0; mt < 2; ++mt) {
#pragma unroll
        for (int t = 0; t < 2; ++t) {
            int col = (ct0 + t) * 16 + l16;
            float bv = bias[col];
#pragma unroll
            for (int r = 0; r < 8; ++r) {
                int row = mbase + mt * 16 + ((lane >> 4) << 3) + r;
                T[(size_t)row * C_ + col] = acc[mt][t][r] + bv;
            }
        }
    }
}

// ---------------------------------------------------------------------------
// Kernel 3: per query  H[16,256] = v[idx[0..15]] @ W0, then
// out[c] = lrelu( max_j H[j,c] + T[n,c] ).
// 8 waves/block, each wave owns 2 column tiles; 4 queries/block so the W0
// B-fragments (L2 resident) are loaded once per K-step and reused 4x.
// A fragments are raw per-lane b64 gathers from the neighbor rows.
// ---------------------------------------------------------------------------
__global__ __launch_bounds__(256) void dga_wmma_kernel(const float* __restrict__ v,
                                                       const float* __restrict__ W,
                                                       const float* __restrict__ T,
                                                       const int* __restrict__ knn,
                                                       float* __restrict__ out) {
    int bid = blockIdx.x;
    int b  = bid / (N_ / QPB);
    int n0 = (bid % (N_ / QPB)) * QPB;
    int tid  = threadIdx.x;
    int lane = tid & 31;
    int wave = tid >> 5;
    int l16  = lane & 15;
    int roff = (lane >> 4) << 1;

    __shared__ int sidx[QPB * K_];
    if (tid < QPB * K_) sidx[tid] = knn[((size_t)b * N_ + n0) * K_ + tid];
    __syncthreads();

    // per-lane gathered neighbor-row base pointers (lane l16 -> neighbor l16)
    const float* rowp[QPB];
#pragma unroll
    for (int qi = 0; qi < QPB; ++qi)
        rowp[qi] = v + ((size_t)b * N_ + sidx[qi * K_ + l16]) * C_;

    int ct0 = wave * 2;

    v8f acc[QPB][2];
#pragma unroll
    for (int qi = 0; qi < QPB; ++qi) {
        acc[qi][0] = (v8f){0.f, 0.f, 0.f, 0.f, 0.f, 0.f, 0.f, 0.f};
        acc[qi][1] = (v8f){0.f, 0.f, 0.f, 0.f, 0.f, 0.f, 0.f, 0.f};
    }

    // K loop over the 256 columns of v (64 steps of 4)
    for (int kk = 0; kk < C_ / 4; ++kk) {
        int krow = kk * 4 + roff;
        const float* wp = W + (size_t)krow * C_ + ct0 * 16 + l16;   // W0 rows
        v2f B0, B1;
        B0.x = wp[0];   B0.y = wp[C_];
        B1.x = wp[16];  B1.y = wp[C_ + 16];
#pragma unroll
        for (int qi = 0; qi < QPB; ++qi) {
            v2f A = *(const v2f*)(rowp[qi] + kk * 4 + roff);        // gather b64
            acc[qi][0] = __builtin_amdgcn_wmma_f32_16x16x4_f32(
                false, A, false, B0, (short)0, acc[qi][0], false, false);
            acc[qi][1] = __builtin_amdgcn_wmma_f32_16x16x4_f32(
                false, A, false, B1, (short)0, acc[qi][1], false, false);
        }
    }

    // ---- epilogue: max over 16 neighbor rows, + T[n,col], leakyReLU(0.2) ----
#pragma unroll
    for (int qi = 0; qi < QPB; ++qi) {
        int n = n0 + qi;
#pragma unroll
        for (int t = 0; t < 2; ++t) {
            float mx = acc[qi][t][0];
#pragma unroll
            for (int r = 1; r < 8; ++r) mx = fmaxf(mx, acc[qi][t][r]);
            // combine lanes L and L+16 (rows 0-7 vs 8-15): ds_swizzle SWAPX16
            float other = __int_as_float(
                __builtin_amdgcn_ds_swizzle(__float_as_int(mx), 0x401F));
            mx = fmaxf(mx, other);
            int col = (ct0 + t) * 16 + l16;
            float val = mx + T[((size_t)b * N_ + n) * C_ + col];
            val = (val > 0.0f) ? val : 0.2f * val;
            if (lane < 16)
                out[((size_t)b * N_ + n) * C_ + col] = val;
        }
    }
}

// ---------------------------------------------------------------------------
extern "C" void kernel_launch(void* const* d_in, const int* in_sizes, int n_in,
                              void* d_out, int out_size, void* d_ws, size_t ws_size,
                              hipStream_t stream) {
    const float* q     = (const float*)d_in[0];   // [B,N,C]
    const float* q_pos = (const float*)d_in[1];   // [B,N,3]
    const float* W     = (const float*)d_in[2];   // [2C,C]
    const float* bias  = (const float*)d_in[3];   // [C]
    // d_in[4] is k (==16), compiled in as K_

    int*   knn = (int*)d_ws;                                   // 2 MB
    float* T   = (float*)((char*)d_ws + (size_t)(2 << 20));    // 32 MB
    float* out = (float*)d_out;                                // [B,N,C]

    knn_kernel<<<(B_ * N_) / 256, 256, 0, stream>>>(q_pos, knn);
    tgemm_kernel<<<(B_ * N_) / MROWS, 256, 0, stream>>>(q, W, bias, T);
    dga_wmma_kernel<<<(B_ * N_) / QPB, 256, 0, stream>>>(/*v=*/q, W, T, knn, out);
}